// GIUNet_74053826117752
// MI455X (gfx1250) — compile-verified
//
#include <hip/hip_runtime.h>
#include <hip/hip_bf16.h>

#define N_NODES 10000
#define N_EDGES 160000
#define F_IN    128
#define MAX_K   512

typedef __attribute__((ext_vector_type(2))) float v2f;
typedef __attribute__((ext_vector_type(8))) float v8f;
typedef __attribute__((ext_vector_type(4))) unsigned int u32x4;
typedef __attribute__((ext_vector_type(8))) int i32x8;
typedef __attribute__((ext_vector_type(4))) int i32x4;

// ---------------------------------------------------------------------------
// Elementwise helpers
// ---------------------------------------------------------------------------
__global__ void zero_f(float* __restrict__ p, long long n) {
    long long i = (long long)blockIdx.x * blockDim.x + threadIdx.x;
    long long stride = (long long)gridDim.x * blockDim.x;
    for (; i < n; i += stride) p[i] = 0.0f;
}

__global__ void ew_add(float* __restrict__ dst, const float* __restrict__ src, long long n) {
    long long i = (long long)blockIdx.x * blockDim.x + threadIdx.x;
    long long stride = (long long)gridDim.x * blockDim.x;
    for (; i < n; i += stride) dst[i] += src[i];
}

// ---------------------------------------------------------------------------
// Edge scatter: agg[dst, :] += x[src, :]  (segment_sum over 160K edges)
// ---------------------------------------------------------------------------
__global__ void scatter_add(const float* __restrict__ x,
                            const long long* __restrict__ src,
                            const long long* __restrict__ dst,
                            float* __restrict__ agg, int D, int nE) {
    long long total = (long long)nE * D;
    long long i = (long long)blockIdx.x * blockDim.x + threadIdx.x;
    long long stride = (long long)gridDim.x * blockDim.x;
    for (; i < total; i += stride) {
        int e = (int)(i / D);
        int f = (int)(i - (long long)e * D);
        long long s = src[e], d = dst[e];
        atomicAdd(&agg[(size_t)d * D + f], x[(size_t)s * D + f]);
    }
}

// ---------------------------------------------------------------------------
// WMMA fp32 GEMM with TDM-staged W panel:
//   OUT[M,N] = act(H[M,K] @ W[K,N] + bias)
// Block = 8 waves; each block owns one 16-col W panel (K x 16, <=32KB) pulled
// into LDS by ONE tensor_load_to_lds DMA, then 8 waves each compute one 16x16
// output tile against it (8 consecutive M tiles).  V_WMMA_F32_16X16X4_F32.
// M%16==0, N%16==0, K%4==0, K<=512 hold for every layer here.
// ---------------------------------------------------------------------------
__global__ __launch_bounds__(256) void gemm_wmma_f32(
        const float* __restrict__ H, const float* __restrict__ W,
        const float* __restrict__ bias, float* __restrict__ OUT,
        int M, int K, int N, int relu) {
    __shared__ float wpanel[MAX_K * 16];          // row r, col c -> wpanel[r*16+c]

    const int tilesN   = N >> 4;
    const int tilesM   = M >> 4;
    const int panel    = blockIdx.x % tilesN;
    const int rowGroup = blockIdx.x / tilesN;
    const int tn       = panel << 4;
    const int lane     = threadIdx.x & 31;
    const int wave     = threadIdx.x >> 5;

    // ---- TDM: DMA the K x 16 panel W[0:K, tn:tn+16] into LDS (wave 0 only;
    //      TDM ignores EXEC, scalar branch gates issue to one wave) ----
    if (wave == 0) {
        unsigned lds_off = (unsigned)(unsigned long long)(const void*)wpanel; // low 32b of flat = LDS offset
        unsigned long long ga = (unsigned long long)(const void*)(W + tn);    // tile start, bytes

        // D# group 0: count=1 | lds_addr | global_addr[56:0] | type=2
        u32x4 g0;
        g0[0] = 1u;                                             // count=1, is_restore=0
        g0[1] = lds_off;                                        // lds_addr (bytes)
        g0[2] = (unsigned)(ga & 0xFFFFFFFFull);                 // global_addr[31:0]
        g0[3] = (unsigned)((ga >> 32) & 0x01FFFFFFull)          // global_addr[56:32]
              | (2u << 30);                                     // type = 2 ("image")

        // D# group 1: data_size=4B; tensor=[N x K], tile=[16 x K], stride0=N
        unsigned td0 = (unsigned)N;            // tensor_dim0 (row length, elems)
        unsigned td1 = (unsigned)K;            // tensor_dim1 (#rows)
        unsigned long long s0 = (unsigned long long)N;  // tensor_dim0_stride
        i32x8 g1;
        g1[0] = (int)(2u << 16);                                // data_size=2 (4B), no mask/flags
        g1[1] = (int)((td0 & 0xFFFFu) << 16);                   // tensor_dim0[15:0] @ bits[63:48]
        g1[2] = (int)((td0 >> 16) | ((td1 & 0xFFFFu) << 16));   // td0[31:16] | td1[15:0]
        g1[3] = (int)((td1 >> 16) | (16u << 16));               // td1[31:16] | tile_dim0=16
        g1[4] = (int)(td1 & 0xFFFFu);                           // tile_dim1=K, tile_dim2=0
        g1[5] = (int)(s0 & 0xFFFFFFFFull);                      // stride0[31:0]
        g1[6] = (int)((s0 >> 32) & 0xFFFFull);                  // stride0[47:32], stride1=0
        g1[7] = 0;

        i32x4 gz4 = {0, 0, 0, 0};                               // 2-D: groups 2/3 unused
        i32x8 gz8 = {0, 0, 0, 0, 0, 0, 0, 0};                   // trailing group (6-arg form)
        __builtin_amdgcn_tensor_load_to_lds(g0, g1, gz4, gz4, gz8, 0);
        __builtin_amdgcn_s_wait_tensorcnt(0);
    }
    __syncthreads();

    // ---- WMMA compute: one 16x16 tile per wave ----
    const int tmTile = rowGroup * 8 + wave;       // wave-uniform
    if (tmTile < tilesM) {
        const int tm   = tmTile << 4;
        const int half = lane >> 4;               // 0 or 1
        const int sub  = lane & 15;
        const int aRow = tm + sub;                // A fragment row (M)
        const int kOff = half * 2;                // K offset within 4-step

        v8f acc = {0.f, 0.f, 0.f, 0.f, 0.f, 0.f, 0.f, 0.f};
        const float* aBase = H + (size_t)aRow * K;
        for (int k = 0; k < K; k += 4) {
            // A: 16x4 fp32 — VGPR0 = K=2*half, VGPR1 = K=2*half+1 (row = lane&15)
            const float* ap = aBase + k + kOff;
            v2f a;
            a.x = ap[0];
            a.y = ap[1];
            __builtin_prefetch(ap + 8, 0, 0);     // global_prefetch_b8 (speculative)
            // B: 4x16 fp32 from LDS panel — rows K=2*half(+1), col = lane&15
            v2f b;
            b.x = wpanel[(k + kOff) * 16 + sub];
            b.y = wpanel[(k + kOff + 1) * 16 + sub];
            acc = __builtin_amdgcn_wmma_f32_16x16x4_f32(
                      false, a, false, b, (short)0, acc, false, false);
        }

        // C/D: VGPR r -> row tm + 8*half + r, col tn + sub
        const int col = tn + sub;
        const float bv = bias ? bias[col] : 0.0f;
        const int rowBase = tm + half * 8;
#pragma unroll
        for (int r = 0; r < 8; ++r) {
            float v = acc[r] + bv;
            if (relu) v = fmaxf(v, 0.0f);
            OUT[(size_t)(rowBase + r) * N + col] = v;
        }
    }
}

// ---------------------------------------------------------------------------
// Column mean over nodes: out[c] = mean_r X[r, c].  One block per column.
// ---------------------------------------------------------------------------
__global__ void colmean(const float* __restrict__ X, float* __restrict__ out,
                        int M, int D) {
    __shared__ float sm[256];
    const int c = blockIdx.x;
    float s = 0.0f;
    for (int r = threadIdx.x; r < M; r += blockDim.x)
        s += X[(size_t)r * D + c];
    sm[threadIdx.x] = s;
    __syncthreads();
    for (int w = blockDim.x >> 1; w > 0; w >>= 1) {
        if ((int)threadIdx.x < w) sm[threadIdx.x] += sm[threadIdx.x + w];
        __syncthreads();
    }
    if (threadIdx.x == 0) out[c] = sm[0] / (float)M;
}

// ---------------------------------------------------------------------------
// Middle MLP on a single 256-vector: mid = relu(p@w1+b1)@w2+b2.  One block.
// ---------------------------------------------------------------------------
__global__ __launch_bounds__(256) void middle_mlp(
        const float* __restrict__ pooled,
        const float* __restrict__ w1, const float* __restrict__ b1,
        const float* __restrict__ w2, const float* __restrict__ b2,
        float* __restrict__ mid) {
    __shared__ float p[256], h[256];
    const int t = threadIdx.x;
    p[t] = pooled[t];
    __syncthreads();
    float s = b1[t];
    for (int k = 0; k < 256; ++k) s += p[k] * w1[k * 256 + t];
    h[t] = fmaxf(s, 0.0f);
    __syncthreads();
    float s2 = b2[t];
    for (int k = 0; k < 256; ++k) s2 += h[k] * w2[k * 256 + t];
    mid[t] = s2;
}

// ---------------------------------------------------------------------------
// Row-wise concat: out[i, :] = [A_row(i) | B[i, :]].  bcastA => A is a single
// row broadcast to every node (the unpooled graph vector).
// ---------------------------------------------------------------------------
__global__ void concat2(const float* __restrict__ A, int dA, int bcastA,
                        const float* __restrict__ B, int dB,
                        float* __restrict__ out, int n) {
    const int D = dA + dB;
    long long total = (long long)n * D;
    long long i = (long long)blockIdx.x * blockDim.x + threadIdx.x;
    long long stride = (long long)gridDim.x * blockDim.x;
    for (; i < total; i += stride) {
        int row = (int)(i / D);
        int j = (int)(i - (long long)row * D);
        out[i] = (j < dA) ? A[(bcastA ? 0 : (size_t)row * dA) + j]
                          : B[(size_t)row * dB + (j - dA)];
    }
}

// ---------------------------------------------------------------------------
// Host-side orchestration
// ---------------------------------------------------------------------------
static inline int grid_for(long long n) {
    long long b = (n + 255) / 256;
    if (b > 65535) b = 65535;   // grid-stride loops cover the rest
    return (int)b;
}

static void launch_gemm(const float* H, const float* W, const float* b, float* O,
                        int M, int K, int N, int relu, hipStream_t s) {
    int tilesM = M >> 4, tilesN = N >> 4;
    int rowGroups = (tilesM + 7) / 8;             // 8 M-tiles (waves) per block
    gemm_wmma_f32<<<rowGroups * tilesN, 256, 0, s>>>(H, W, b, O, M, K, N, relu);
}

// One GIN layer: out = MLP(x + segment_sum(x[src], dst))
static void gin_layer(const float* X, int D,
                      const long long* src, const long long* dst,
                      const float* w1, const float* b1, int Dh,
                      const float* w2, const float* b2, int Do,
                      float* agg, float* hid, float* out, hipStream_t s) {
    long long n = (long long)N_NODES * D;
    zero_f<<<grid_for(n), 256, 0, s>>>(agg, n);
    scatter_add<<<4096, 256, 0, s>>>(X, src, dst, agg, D, N_EDGES);
    ew_add<<<grid_for(n), 256, 0, s>>>(agg, X, n);
    launch_gemm(agg, w1, b1, hid, N_NODES, D, Dh, 1, s);
    launch_gemm(hid, w2, b2, out, N_NODES, Dh, Do, 0, s);
}

extern "C" void kernel_launch(void* const* d_in, const int* in_sizes, int n_in,
                              void* d_out, int out_size, void* d_ws, size_t ws_size,
                              hipStream_t stream) {
    (void)in_sizes; (void)n_in; (void)out_size; (void)ws_size;

    // ---- inputs (setup_inputs dict order; params nested dicts flatten in
    //      insertion order: conv1..3, middle, up1..3, each w1,b1,w2,b2) ----
    const float*     x   = (const float*)d_in[0];
    const long long* ei  = (const long long*)d_in[1];     // int64 [2, E]
    const long long* src = ei;
    const long long* dst = ei + N_EDGES;

    const float** P = (const float**)(d_in + 2);          // 28 param tensors

    // ---- workspace carve (fp32) ----
    float* ws     = (float*)d_ws;
    float* agg    = ws;                                   // N*512
    float* cat    = agg + (size_t)N_NODES * 512;          // N*512
    float* hid    = cat + (size_t)N_NODES * 512;          // N*256
    float* x1     = hid + (size_t)N_NODES * 256;          // N*64
    float* x2     = x1  + (size_t)N_NODES * 64;           // N*128
    float* x3     = x2  + (size_t)N_NODES * 128;          // N*256
    float* u1     = x3  + (size_t)N_NODES * 256;          // N*128
    float* u2     = u1  + (size_t)N_NODES * 128;          // N*64
    float* pooled = u2  + (size_t)N_NODES * 64;           // 256
    float* mid    = pooled + 256;                         // 256

    // ---- encoder ----
    gin_layer(x,  F_IN, src, dst, P[0],  P[1],  64,  P[2],  P[3],  64,  agg, hid, x1, stream);
    gin_layer(x1, 64,   src, dst, P[4],  P[5],  128, P[6],  P[7],  128, agg, hid, x2, stream);
    gin_layer(x2, 128,  src, dst, P[8],  P[9],  256, P[10], P[11], 256, agg, hid, x3, stream);

    // ---- global mean pool + middle MLP ----
    colmean<<<256, 256, 0, stream>>>(x3, pooled, N_NODES, 256);
    middle_mlp<<<1, 256, 0, stream>>>(pooled, P[12], P[13], P[14], P[15], mid);

    // ---- decoder with skips ----
    // up1: [broadcast(mid) | x3]  (512) -> 128
    concat2<<<grid_for((long long)N_NODES * 512), 256, 0, stream>>>(mid, 256, 1, x3, 256, cat, N_NODES);
    gin_layer(cat, 512, src, dst, P[16], P[17], 128, P[18], P[19], 128, agg, hid, u1, stream);

    // up2: [u1 | x2]  (256) -> 64
    concat2<<<grid_for((long long)N_NODES * 256), 256, 0, stream>>>(u1, 128, 0, x2, 128, cat, N_NODES);
    gin_layer(cat, 256, src, dst, P[20], P[21], 64, P[22], P[23], 64, agg, hid, u2, stream);

    // up3: [u2 | x]  (192) -> 16   (writes d_out directly)
    concat2<<<grid_for((long long)N_NODES * 192), 256, 0, stream>>>(u2, 64, 0, x, F_IN, cat, N_NODES);
    gin_layer(cat, 192, src, dst, P[24], P[25], 16, P[26], P[27], 16, agg, hid, (float*)d_out, stream);
}